// AttentionLayer_50543175139752
// MI455X (gfx1250) — compile-verified
//
#include <hip/hip_runtime.h>
#include <hip/hip_bf16.h>

// ---------------------------------------------------------------------------
// AttentionLayer for MI455X (gfx1250).
//  - All GEMMs on the f32 WMMA pipe (V_WMMA_F32_16X16X4_F32, exact fp32).
//  - B/Q/S LDS panels stored k-pair interleaved -> 1x ds_load_b64 per fragment,
//    pair-row pitch 80 v2f (row offset == 32 banks) -> conflict-free.
//  - Conv A-panel staged by the Tensor Data Mover (tensor_load_to_lds) with
//    LDS padding (128 DW + 8 DW pad -> [32][136]), double buffered and
//    overlapped with WMMA compute; s_wait_tensorcnt + barrier per chunk.
// ---------------------------------------------------------------------------

typedef __attribute__((ext_vector_type(2))) float v2f;
typedef __attribute__((ext_vector_type(8))) float v8f;
typedef __attribute__((ext_vector_type(4))) unsigned int u32x4;
typedef __attribute__((ext_vector_type(8))) int i32x8;
typedef __attribute__((ext_vector_type(4))) int i32x4;

#define B_    8
#define CIN   256
#define HW    4096
#define WID   64
#define CK    64
#define LL    256
#define COUT  256
#define CCAT  320   // Cin + Ck

__device__ __forceinline__ v8f wmma_f32(v2f a, v2f b, v8f c) {
  return __builtin_amdgcn_wmma_f32_16x16x4_f32(false, a, false, b, (short)0, c,
                                               false, false);
}

#if __has_builtin(__builtin_amdgcn_tensor_load_to_lds) && \
    __has_builtin(__builtin_amdgcn_s_wait_tensorcnt)
#define HAVE_TDM 1
#else
#define HAVE_TDM 0
#endif

#if HAVE_TDM
// TDM load of a 2D tile: 32 rows x 128 floats, row stride COUT floats, into
// LDS at lds_byte with 8-DW padding every 128 DW -> [32][136] float layout.
__device__ __forceinline__ void tdm_load_w2_panel(const float* src,
                                                  unsigned lds_byte) {
  unsigned long long ga = (unsigned long long)(size_t)src;
  u32x4 g0;
  g0[0] = 1u;                                   // count=1 (valid user D#)
  g0[1] = lds_byte;                             // lds_addr
  g0[2] = (unsigned)(ga & 0xffffffffull);       // global_addr[31:0]
  g0[3] = (unsigned)((ga >> 32) & 0x01ffffffu)  // global_addr[56:32]
          | (2u << 30);                         // type = 2 ("image")
  i32x8 g1;
  g1[0] = (2 << 16)      // data_size = 4B
        | (1 << 20)      // pad_enable
        | (6 << 22)      // pad_interval: 128 DWORDs
        | (7 << 25);     // pad_amount:   8 DWORDs
  g1[1] = (COUT & 0xffff) << 16;                        // tensor_dim0 lo16
  g1[2] = ((2880 & 0xffff) << 16) | ((COUT >> 16) & 0xffff); // dim1 lo|dim0 hi
  g1[3] = (128 << 16) | ((2880 >> 16) & 0xffff);        // tile_dim0 | dim1 hi
  g1[4] = 32;                                           // tile_dim1 (dim2=0)
  g1[5] = COUT;                                         // tensor_dim0_stride
  g1[6] = 0;
  g1[7] = 0;
  i32x4 gz4 = {0, 0, 0, 0};
  i32x8 gz8 = {0, 0, 0, 0, 0, 0, 0, 0};
  // 6-arg form on this toolchain: (g0, g1, g2, g3, g4, cpol)
  __builtin_amdgcn_tensor_load_to_lds(g0, g1, gz4, gz4, gz8, 0);
}
#endif

// ---------------------------------------------------------------------------
// Kernel 0: repack W2 [Cout][320][3][3] -> W2t [rs][c][Cout]
// ---------------------------------------------------------------------------
__global__ void repack_w2_kernel(const float* __restrict__ W2,
                                 float* __restrict__ W2t) {
  int idx = blockIdx.x * blockDim.x + threadIdx.x;   // [0, 9*320*256)
  if (idx >= 9 * CCAT * COUT) return;
  int co = idx % COUT;
  int t  = idx / COUT;
  int c  = t % CCAT;
  int rs = t / CCAT;
  W2t[idx] = W2[((size_t)co * CCAT + c) * 9 + rs];
}

// ---------------------------------------------------------------------------
// Kernel 1: fused per-pixel attention. 128 threads (4 waves) per 64-px tile.
// Q/S panels in LDS use k-pair interleaved layout: pair-row = 80 v2f (160 f).
// ---------------------------------------------------------------------------
__global__ __launch_bounds__(128) void attn_kernel(
    const float* __restrict__ img, const float* __restrict__ vemb,
    const float* __restrict__ W1, const float* __restrict__ b1,
    float* __restrict__ value_ws) {
  __shared__ float Qsf[(CK / 2) * 160];   // [kpair][px][2]
  __shared__ float Ssf[(LL / 2) * 160];   // [lpair][px][2]
  __shared__ float inv_sum[64];

  const int tid  = threadIdx.x;
  const int lane = tid & 31;
  const int wave = tid >> 5;       // 0..3
  const int lo   = lane & 15;
  const int hi   = lane >> 4;      // 0/1
  const int b    = blockIdx.x >> 6;
  const int p0   = (blockIdx.x & 63) * 64;

  const float* imgb = img + (size_t)b * CIN * HW;
  const float* vb   = vemb + (size_t)b * CK * LL;

  // ---------------- Phase 1: Q = W1 @ X + b1 ----------------
  {
    const int m0 = 16 * wave;
    v8f acc[4] = {};
    for (int k0 = 0; k0 < CIN; k0 += 4) {
      v2f a;
      const float* ap = W1 + (size_t)(m0 + lo) * CIN + k0 + 2 * hi;
      a.x = ap[0];
      a.y = ap[1];
      for (int nt = 0; nt < 4; ++nt) {
        v2f bf;
        const float* bp = imgb + (size_t)(k0 + 2 * hi) * HW + p0 + nt * 16 + lo;
        bf.x = bp[0];
        bf.y = bp[HW];
        acc[nt] = wmma_f32(a, bf, acc[nt]);
      }
    }
    for (int nt = 0; nt < 4; ++nt)
      for (int r = 0; r < 8; ++r) {
        int m  = m0 + r + 8 * hi;
        int px = nt * 16 + lo;
        Qsf[(m >> 1) * 160 + 2 * px + (m & 1)] = acc[nt][r] + b1[m];
      }
  }
  __syncthreads();

  // ---------------- Phase 2: S = V^T @ Q ----------------
  for (int mt = 0; mt < 4; ++mt) {
    const int l0 = 64 * wave + 16 * mt;
    v8f acc[4] = {};
    for (int k0 = 0; k0 < CK; k0 += 4) {
      v2f a;  // A[l][k] = vemb[k][l]
      const float* ap = vb + (size_t)(k0 + 2 * hi) * LL + l0 + lo;
      a.x = ap[0];
      a.y = ap[LL];
      for (int nt = 0; nt < 4; ++nt) {
        v2f bf = *(const v2f*)&Qsf[(k0 / 2 + hi) * 160 + 2 * (nt * 16 + lo)];
        acc[nt] = wmma_f32(a, bf, acc[nt]);
      }
    }
    for (int nt = 0; nt < 4; ++nt)
      for (int r = 0; r < 8; ++r) {
        int l  = l0 + r + 8 * hi;
        int px = nt * 16 + lo;
        Ssf[(l >> 1) * 160 + 2 * px + (l & 1)] = acc[nt][r];
      }
  }
  __syncthreads();

  // ---------------- softmax over L per pixel column ----------------
  if (tid < 64) {
    float mx = -3.402823466e38f;
    for (int l = 0; l < LL; ++l)
      mx = fmaxf(mx, Ssf[(l >> 1) * 160 + 2 * tid + (l & 1)]);
    float sum = 0.f;
    for (int l = 0; l < LL; ++l) {
      int idx = (l >> 1) * 160 + 2 * tid + (l & 1);
      float e = __expf(Ssf[idx] - mx);
      Ssf[idx] = e;
      sum += e;
    }
    inv_sum[tid] = 1.f / sum;
  }
  __syncthreads();

  // ---------------- Phase 3: value = V @ exp(S), scaled ----------------
  {
    const int m0 = 16 * wave;
    v8f acc[4] = {};
    for (int k0 = 0; k0 < LL; k0 += 4) {
      v2f a;  // A[c][l] = vemb[c][l]
      const float* ap = vb + (size_t)(m0 + lo) * LL + k0 + 2 * hi;
      a.x = ap[0];
      a.y = ap[1];
      for (int nt = 0; nt < 4; ++nt) {
        v2f bf = *(const v2f*)&Ssf[(k0 / 2 + hi) * 160 + 2 * (nt * 16 + lo)];
        acc[nt] = wmma_f32(a, bf, acc[nt]);
      }
    }
    float* vout = value_ws + (size_t)b * CK * HW;
    for (int nt = 0; nt < 4; ++nt) {
      int px = nt * 16 + lo;
      float is = inv_sum[px];
      for (int r = 0; r < 8; ++r) {
        int c = m0 + r + 8 * hi;
        vout[(size_t)c * HW + p0 + px] = acc[nt][r] * is;
      }
    }
  }
}

// ---------------------------------------------------------------------------
// Kernel 2: 3x3 conv as implicit GEMM, double-buffered.
// A panel [32][136] filled by the TDM; B panel k-pair interleaved [16][80] v2f
// gathered (im2col + zero pad) into registers and stored while computing.
// 256 threads = 8 waves (2 M x 4 N); wave tile 64(M) x 16(N).
// ---------------------------------------------------------------------------
__global__ __launch_bounds__(256) void conv_kernel(
    const float* __restrict__ img, const float* __restrict__ value_ws,
    const float* __restrict__ W2t, const float* __restrict__ b2,
    float* __restrict__ out) {
  __shared__ float As[2][32][136];      // [buf][k][m]
  __shared__ float Bsf[2][16 * 160];    // [buf][kpair][x][2]

  const int tid  = threadIdx.x;
  const int lane = tid & 31;
  const int wave = tid >> 5;     // 0..7
  const int lo   = lane & 15;
  const int hi   = lane >> 4;
  const int wm   = wave >> 2;    // 0..1
  const int wn   = wave & 3;     // 0..3

  const int b       = blockIdx.z;
  const int co_base = blockIdx.y * 128;
  const int y       = blockIdx.x;          // image row

  const float* imgb = img + (size_t)b * CIN * HW;
  const float* valb = value_ws + (size_t)b * CK * HW;

  const int mcol = wm * 64 + lo;           // + mt*16 added in loop
  const int n16  = wn * 16 + lo;

  v8f acc[4] = {};
  float breg[8];

  const int NC = 9 * (CCAT / 32);          // 90 K-chunks

  // ---- gather B for a chunk into registers ----
  auto gatherB = [&](int ch, float* reg) {
    const int rs = ch / (CCAT / 32);
    const int cc = ch % (CCAT / 32);
    const int r = rs / 3, s = rs % 3;
    const int yy = y + r - 1;
    const bool row_ok = (yy >= 0) && (yy < WID);
    const int c0 = cc * 32;
    for (int i = 0; i < 8; ++i) {
      int idx = tid + 256 * i;
      int kk  = idx >> 6;
      int x   = idx & 63;
      int xx  = x + s - 1;
      float v = 0.f;
      if (row_ok && xx >= 0 && xx < WID) {
        int c = c0 + kk;
        const float* plane = (c < CIN) ? (imgb + (size_t)c * HW)
                                       : (valb + (size_t)(c - CIN) * HW);
        v = plane[yy * WID + xx];
      }
      reg[i] = v;
    }
  };
  auto storeB = [&](int nb, const float* reg) {
    for (int i = 0; i < 8; ++i) {
      int idx = tid + 256 * i;
      int kk  = idx >> 6;
      int x   = idx & 63;
      Bsf[nb][(kk >> 1) * 160 + 2 * x + (kk & 1)] = reg[i];
    }
  };
  auto loadA = [&](int ch, int nb) {
    const int rs = ch / (CCAT / 32);
    const int c0 = (ch % (CCAT / 32)) * 32;
    const float* src = W2t + ((size_t)rs * CCAT + c0) * COUT + co_base;
#if HAVE_TDM
    if (wave == 0)
      tdm_load_w2_panel(src, (unsigned)(size_t)&As[nb][0][0]);
#else
    for (int i = 0; i < 16; ++i) {
      int idx = tid + 256 * i;
      int kk  = idx >> 7;
      int m   = idx & 127;
      As[nb][kk][m] = src[(size_t)kk * COUT + m];
    }
#endif
  };

  // ---- prologue: chunk 0 into buffer 0 ----
  loadA(0, 0);
  gatherB(0, breg);
  storeB(0, breg);
#if HAVE_TDM
  if (wave == 0) __builtin_amdgcn_s_wait_tensorcnt(0);
#endif
  __syncthreads();

  for (int ch = 0; ch < NC; ++ch) {
    const int buf = ch & 1;
    const int nb  = buf ^ 1;
    if (ch + 1 < NC) {
      loadA(ch + 1, nb);                 // TDM (wave 0) -> As[nb]
      gatherB(ch + 1, breg);             // global -> regs
      storeB(nb, breg);                  // regs -> Bsf[nb]
    }
    // ---- compute chunk ch from buffer buf ----
    const float* Asb = &As[buf][0][0];
    const float* Bsb = &Bsf[buf][0];
    for (int kk = 0; kk < 32; kk += 4) {
      v2f bf = *(const v2f*)&Bsb[(kk / 2 + hi) * 160 + 2 * n16];
      for (int mt = 0; mt < 4; ++mt) {
        v2f a;
        a.x = Asb[(kk + 2 * hi) * 136 + mcol + mt * 16];
        a.y = Asb[(kk + 2 * hi + 1) * 136 + mcol + mt * 16];
        acc[mt] = wmma_f32(a, bf, acc[mt]);
      }
    }
#if HAVE_TDM
    if (wave == 0 && ch + 1 < NC) __builtin_amdgcn_s_wait_tensorcnt(0);
#endif
    __syncthreads();
  }

  float* outb = out + (size_t)b * COUT * HW;
  for (int mt = 0; mt < 4; ++mt) {
    for (int rr = 0; rr < 8; ++rr) {
      int co = co_base + wm * 64 + mt * 16 + rr + 8 * hi;
      outb[(size_t)co * HW + y * WID + n16] = acc[mt][rr] + b2[co];
    }
  }
}

// ---------------------------------------------------------------------------
extern "C" void kernel_launch(void* const* d_in, const int* in_sizes, int n_in,
                              void* d_out, int out_size, void* d_ws,
                              size_t ws_size, hipStream_t stream) {
  const float* img  = (const float*)d_in[0];
  const float* vemb = (const float*)d_in[1];
  const float* W1   = (const float*)d_in[2];
  const float* b1   = (const float*)d_in[3];
  const float* W2   = (const float*)d_in[4];
  const float* b2   = (const float*)d_in[5];
  float* out = (float*)d_out;

  // workspace: value [8][64][4096] f32 (8 MB) then W2t [9][320][256] (2.95 MB)
  float* value_ws = (float*)d_ws;
  float* W2t      = value_ws + (size_t)B_ * CK * HW;

  repack_w2_kernel<<<(9 * CCAT * COUT + 255) / 256, 256, 0, stream>>>(W2, W2t);
  attn_kernel<<<B_ * (HW / 64), 128, 0, stream>>>(img, vemb, W1, b1, value_ws);
  dim3 grid(WID, COUT / 128, B_);
  conv_kernel<<<grid, 256, 0, stream>>>(img, value_ws, W2t, b2, out);
}